// GATV1Module_81226421502294
// MI455X (gfx1250) — compile-verified
//
#include <hip/hip_runtime.h>
#include <hip/hip_bf16.h>

typedef float v2f __attribute__((ext_vector_type(2)));
typedef float v8f __attribute__((ext_vector_type(8)));

#define GAT_DIM 64
#define NEG_SLOPE 0.2f
#define THRESH 1.0f

// ---------------------------------------------------------------------------
// order-preserving float <-> uint encoding for atomic max on floats
// ---------------------------------------------------------------------------
__device__ __forceinline__ unsigned enc_f(float f) {
    unsigned u = __float_as_uint(f);
    return (u >> 31) ? ~u : (u | 0x80000000u);
}
__device__ __forceinline__ float dec_f(unsigned k) {
    return __uint_as_float((k >> 31) ? (k ^ 0x80000000u) : ~k);
}
#define ENC_NEG_INF 0x007FFFFFu   // enc_f(-inf)

__device__ __forceinline__ float leaky(float v) {
    return v > 0.0f ? v : NEG_SLOPE * v;
}

__device__ __forceinline__ void edge_pair(const int* __restrict__ ei, int e,
                                          int nE, int& s, int& d) {
    if (e < nE) { s = ei[e]; d = ei[nE + e]; }
    else        { s = e - nE; d = s; }        // appended self-loop
}

// ---------------------------------------------------------------------------
// K1: zero 'out' accumulator, init segment-max keys and segment sums
// ---------------------------------------------------------------------------
__global__ void init_kernel(float* __restrict__ out, unsigned* __restrict__ mkey,
                            float* __restrict__ sbuf, int n, int n64) {
    int i = blockIdx.x * blockDim.x + threadIdx.x;
    if (i < n64) out[i] = 0.0f;
    if (i < n) { mkey[i] = ENC_NEG_INF; sbuf[i] = 0.0f; }
}

// ---------------------------------------------------------------------------
// K2: hw = h @ W   (fp32 WMMA 16x16x4, W staged in LDS)
// block = 128 threads (4 waves); each wave owns a 16-node M-tile and all
// four 16-wide N-tiles; A fragment is reused across the 4 WMMAs per k-step.
// ---------------------------------------------------------------------------
__global__ __launch_bounds__(128) void gemm_wmma_kernel(
        const float* __restrict__ h, const float* __restrict__ W,
        float* __restrict__ hw, int nnodes) {
    __shared__ float Wl[GAT_DIM * GAT_DIM];   // 16 KB
    int tid = threadIdx.x;
    #pragma unroll
    for (int i = 0; i < 32; ++i) Wl[tid + i * 128] = W[tid + i * 128];
    __syncthreads();

    int wave = tid >> 5;
    int lane = tid & 31;
    int mbase = blockIdx.x * 64 + wave * 16;
    if (mbase >= nnodes) return;

    int col  = lane & 15;            // N (and M-row) index within tile
    int koff = (lane < 16) ? 0 : 2;  // K pair selector per lane half
    const float* hrow = h + (size_t)(mbase + col) * GAT_DIM;

    v8f acc0 = {}, acc1 = {}, acc2 = {}, acc3 = {};
    #pragma unroll
    for (int kk = 0; kk < 16; ++kk) {
        int k0 = kk * 4;
        v2f a;
        a.x = hrow[k0 + koff];
        a.y = hrow[k0 + koff + 1];
        const float* w0 = &Wl[(k0 + koff) * GAT_DIM + col];
        const float* w1 = &Wl[(k0 + koff + 1) * GAT_DIM + col];
        v2f b0, b1, b2, b3;
        b0.x = w0[0];  b0.y = w1[0];
        b1.x = w0[16]; b1.y = w1[16];
        b2.x = w0[32]; b2.y = w1[32];
        b3.x = w0[48]; b3.y = w1[48];
        acc0 = __builtin_amdgcn_wmma_f32_16x16x4_f32(false, a, false, b0, (short)0, acc0, false, false);
        acc1 = __builtin_amdgcn_wmma_f32_16x16x4_f32(false, a, false, b1, (short)0, acc1, false, false);
        acc2 = __builtin_amdgcn_wmma_f32_16x16x4_f32(false, a, false, b2, (short)0, acc2, false, false);
        acc3 = __builtin_amdgcn_wmma_f32_16x16x4_f32(false, a, false, b3, (short)0, acc3, false, false);
    }

    // C/D layout: VGPR r -> M = r (lanes 0-15) or M = r+8 (lanes 16-31)
    int row_out = mbase + ((lane < 16) ? 0 : 8);
    #pragma unroll
    for (int r = 0; r < 8; ++r) {
        float* dst = hw + (size_t)(row_out + r) * GAT_DIM + col;
        dst[0]  = acc0[r];
        dst[16] = acc1[r];
        dst[32] = acc2[r];
        dst[48] = acc3[r];
    }
}

// ---------------------------------------------------------------------------
// K3: per-node attention logits  alpha_s[i] = hw[i,:]·a_src, alpha_d likewise
// one wave per node, 2 elements per lane, shfl_xor reduction
// ---------------------------------------------------------------------------
__global__ void alpha_kernel(const float* __restrict__ hw,
                             const float* __restrict__ a_src,
                             const float* __restrict__ a_dst,
                             float* __restrict__ as, float* __restrict__ ad,
                             int nnodes) {
    int node = blockIdx.x * (blockDim.x >> 5) + (threadIdx.x >> 5);
    int lane = threadIdx.x & 31;
    if (node >= nnodes) return;
    const float* r = hw + (size_t)node * GAT_DIM;
    float h0 = r[lane], h1 = r[lane + 32];
    float ps = h0 * a_src[lane] + h1 * a_src[lane + 32];
    float pd = h0 * a_dst[lane] + h1 * a_dst[lane + 32];
    #pragma unroll
    for (int off = 16; off > 0; off >>= 1) {
        ps += __shfl_xor(ps, off, 32);
        pd += __shfl_xor(pd, off, 32);
    }
    if (lane == 0) { as[node] = ps; ad[node] = pd; }
}

// ---------------------------------------------------------------------------
// K4: segment max of leaky-relu edge logits over dst
// ---------------------------------------------------------------------------
__global__ void edge_max_kernel(const int* __restrict__ ei,
                                const float* __restrict__ as,
                                const float* __restrict__ ad,
                                unsigned* __restrict__ mkey, int nE, int nN) {
    int e = blockIdx.x * blockDim.x + threadIdx.x;
    if (e >= nE + nN) return;
    int s, d; edge_pair(ei, e, nE, s, d);
    float v = leaky(as[s] + ad[d]);
    atomicMax(mkey + d, enc_f(v));
}

// ---------------------------------------------------------------------------
// K5: segment sum of exp(e - m[dst])
// ---------------------------------------------------------------------------
__global__ void edge_sum_kernel(const int* __restrict__ ei,
                                const float* __restrict__ as,
                                const float* __restrict__ ad,
                                const unsigned* __restrict__ mkey,
                                float* __restrict__ sbuf, int nE, int nN) {
    int e = blockIdx.x * blockDim.x + threadIdx.x;
    if (e >= nE + nN) return;
    int s, d; edge_pair(ei, e, nE, s, d);
    float v = leaky(as[s] + ad[d]);
    atomicAdd(sbuf + d, __expf(v - dec_f(mkey[d])));
}

// ---------------------------------------------------------------------------
// K6: out[dst,:] += attn * hw[src,:]   — one wave per edge, float2 per lane
// ---------------------------------------------------------------------------
__global__ void edge_agg_kernel(const int* __restrict__ ei,
                                const float* __restrict__ as,
                                const float* __restrict__ ad,
                                const unsigned* __restrict__ mkey,
                                const float* __restrict__ sbuf,
                                const float* __restrict__ hw,
                                float* __restrict__ out, int nE, int nN) {
    int wid  = blockIdx.x * (blockDim.x >> 5) + (threadIdx.x >> 5);
    int lane = threadIdx.x & 31;
    if (wid >= nE + nN) return;
    int s, d; edge_pair(ei, wid, nE, s, d);
    float v    = leaky(as[s] + ad[d]);
    float attn = __expf(v - dec_f(mkey[d])) / sbuf[d];
    const float2* hp = (const float2*)(hw + (size_t)s * GAT_DIM);
    float2 hv = hp[lane];
    float* o = out + (size_t)d * GAT_DIM + 2 * lane;
    atomicAdd(o,     attn * hv.x);
    atomicAdd(o + 1, attn * hv.y);
}

// ---------------------------------------------------------------------------
// K7: h_next = clamp(out + b, -1, 1)
// ---------------------------------------------------------------------------
__global__ void finalize_kernel(const float* __restrict__ out,
                                const float* __restrict__ b,
                                float* __restrict__ hnext, int n64) {
    int i = blockIdx.x * blockDim.x + threadIdx.x;
    if (i >= n64) return;
    float v = out[i] + b[i & (GAT_DIM - 1)];
    v = fminf(THRESH, fmaxf(-THRESH, v));
    hnext[i] = v;
}

// ---------------------------------------------------------------------------
extern "C" void kernel_launch(void* const* d_in, const int* in_sizes, int n_in,
                              void* d_out, int out_size, void* d_ws, size_t ws_size,
                              hipStream_t stream) {
    const float* x     = (const float*)d_in[0];
    const int*   ei    = (const int*)d_in[1];
    const float* W     = (const float*)d_in[2];
    const float* a_src = (const float*)d_in[3];
    const float* a_dst = (const float*)d_in[4];
    const float* bias  = (const float*)d_in[5];
    float* outF = (float*)d_out;

    const int N = in_sizes[0] / GAT_DIM;
    const int E = in_sizes[1] / 2;
    const int n64 = N * GAT_DIM;
    const int tot = E + N;          // edges + self loops

    char* ws = (char*)d_ws;
    float*    hwb  = (float*)ws;                                  // N*64
    float*    outb = (float*)(ws + (size_t)n64 * 4);              // N*64
    float*    as   = (float*)(ws + (size_t)n64 * 8);              // N
    float*    ad   = as + N;                                      // N
    unsigned* mkey = (unsigned*)(ad + N);                         // N
    float*    sbuf = (float*)(mkey + N);                          // N

    for (int l = 0; l < 3; ++l) {
        const float* h = (l == 0) ? x : outF;   // d_out ping-pongs as feature buf
        init_kernel<<<(n64 + 255) / 256, 256, 0, stream>>>(outb, mkey, sbuf, N, n64);
        gemm_wmma_kernel<<<(N + 63) / 64, 128, 0, stream>>>(h, W + l * GAT_DIM * GAT_DIM, hwb, N);
        alpha_kernel<<<(N + 7) / 8, 256, 0, stream>>>(hwb, a_src + l * GAT_DIM,
                                                      a_dst + l * GAT_DIM, as, ad, N);
        edge_max_kernel<<<(tot + 255) / 256, 256, 0, stream>>>(ei, as, ad, mkey, E, N);
        edge_sum_kernel<<<(tot + 255) / 256, 256, 0, stream>>>(ei, as, ad, mkey, sbuf, E, N);
        edge_agg_kernel<<<(tot + 7) / 8, 256, 0, stream>>>(ei, as, ad, mkey, sbuf,
                                                           hwb, outb, E, N);
        finalize_kernel<<<(n64 + 255) / 256, 256, 0, stream>>>(outb, bias + l * GAT_DIM,
                                                               outF, n64);
    }
}